// ExtractRelevantPatchesLayer_68521908240694
// MI455X (gfx1250) — compile-verified
//
#include <hip/hip_runtime.h>

#define PATCH   64
#define NPTCH   21
#define GRID_   7
#define BATCH   64
#define DIM     448
#define POOLS   49   // 7*7

typedef float        v2f   __attribute__((ext_vector_type(2)));
typedef float        v8f   __attribute__((ext_vector_type(8)));
typedef unsigned int u32x4 __attribute__((ext_vector_type(4)));
typedef int          i32x4 __attribute__((ext_vector_type(4)));
typedef int          i32x8 __attribute__((ext_vector_type(8)));

// ---------------------------------------------------------------------------
// Kernel 1: 64x64 non-overlapping average pool.
// One block per pool (B*49 = 3136 blocks, 256 threads = 8 waves).
// Each thread sums 16 elements into two partials; a wave's 64 partials fill
// the 4x16 B operand of V_WMMA_F32_16X16X4_F32 with A = ones, so D row 0
// holds 16 column sums (layout-independent since A is all ones). A width-16
// butterfly finishes the wave sum; LDS combines the 8 wave sums.
// ---------------------------------------------------------------------------
__global__ __launch_bounds__(256) void pool_kernel(const float* __restrict__ hm,
                                                   float* __restrict__ avg) {
  int blk  = blockIdx.x;          // b*49 + pool
  int b    = blk / POOLS;
  int pool = blk % POOLS;
  int gy   = pool / GRID_;
  int gx   = pool % GRID_;
  int t    = threadIdx.x;
  int c    = t & 63;              // column within patch
  int r0   = t >> 6;              // 0..3 base row

  const float* base = hm + (size_t)(b * DIM + gy * PATCH) * DIM + gx * PATCH;

  float p0 = 0.f, p1 = 0.f;
#pragma unroll
  for (int i = 0; i < 16; ++i) {
    float v = base[(size_t)(r0 + 4 * i) * DIM + c];
    if (i & 1) p1 += v; else p0 += v;
  }

  float s;
#if __has_builtin(__builtin_amdgcn_wmma_f32_16x16x4_f32)
  v2f bmat;  bmat[0] = p0;  bmat[1] = p1;     // wave's 64 partials = full B
  v2f aones; aones[0] = 1.f; aones[1] = 1.f;  // ones: layout-independent
  v8f cz = {};
  v8f d = __builtin_amdgcn_wmma_f32_16x16x4_f32(
      false, aones, false, bmat, (short)0, cz, false, false);
  s = d[0];  // lanes 0-15: column sums (row M=0); lanes 16-31 duplicate (M=8)
#else
  s = p0 + p1;
  s += __shfl_xor(s, 16, 32);
#endif
  // reduce the 16 column sums across lanes (width-16 butterfly)
  s += __shfl_xor(s, 1, 16);
  s += __shfl_xor(s, 2, 16);
  s += __shfl_xor(s, 4, 16);
  s += __shfl_xor(s, 8, 16);

  __shared__ float wsum[8];
  int wave = t >> 5;
  if ((t & 31) == 0) wsum[wave] = s;
  __syncthreads();
  if (t == 0) {
    float tot = 0.f;
#pragma unroll
    for (int w = 0; w < 8; ++w) tot += wsum[w];
    avg[blk] = tot * (1.0f / 4096.0f);
  }
}

// ---------------------------------------------------------------------------
// Kernel 2: per-batch top-21 of the 49 pooled values (descending, stable
// ties -> lower index first, matching jax.lax.top_k). Rank by counting.
// ---------------------------------------------------------------------------
__global__ __launch_bounds__(64) void topk_kernel(const float* __restrict__ avg,
                                                  int* __restrict__ idx) {
  int b = blockIdx.x;
  int t = threadIdx.x;
  __shared__ float v[POOLS];
  if (t < POOLS) v[t] = avg[b * POOLS + t];
  __syncthreads();
  if (t < POOLS) {
    float mv = v[t];
    int rank = 0;
    for (int j = 0; j < POOLS; ++j) {
      float u = v[j];
      rank += (u > mv) || (u == mv && j < t);
    }
    if (rank < NPTCH) idx[b * NPTCH + rank] = t;
  }
}

// ---------------------------------------------------------------------------
// Kernel 3: gather selected patches. One wave per patch (1344 blocks).
// Each patch is a 2D tile: 192 f32 per row x 64 rows, src row stride 1344
// elements, dst contiguous. This is exactly a TDM descriptor, so we stage
// the 48KB tile through LDS with tensor_load_to_lds / tensor_store_from_lds
// (TENSORcnt-synchronized). Fallback: vectorized float4 copy.
// ---------------------------------------------------------------------------
__global__ __launch_bounds__(32) void gather_kernel(const float* __restrict__ image,
                                                    const int* __restrict__ idx,
                                                    float* __restrict__ out) {
  __shared__ float buf[PATCH * PATCH * 3];  // 48 KB tile staging
  int patch = blockIdx.x;                   // 0..1343 (b*21 + k, rank order)
  int b = patch / NPTCH;
  int p = idx[patch];                       // pool index 0..48
  int gy = p / GRID_;
  int gx = p % GRID_;

  unsigned long long srcElem =
      (unsigned long long)(b * DIM + gy * PATCH) * DIM + (unsigned long long)(gx * PATCH);
  unsigned long long srcAddr = (unsigned long long)(uintptr_t)image + srcElem * 3ull * 4ull;
  unsigned long long dstAddr = (unsigned long long)(uintptr_t)out +
                               (unsigned long long)patch * (PATCH * PATCH * 3ull * 4ull);

#if __has_builtin(__builtin_amdgcn_tensor_load_to_lds) && \
    __has_builtin(__builtin_amdgcn_tensor_store_from_lds)
  unsigned int lds = (unsigned int)(uintptr_t)(void*)buf;  // low 32b = LDS offset

  const unsigned int W  = PATCH * 3;      // 192 elements per tile row
  const unsigned int H  = PATCH;          // 64 rows
  const unsigned int SS = DIM * 3;        // src row stride, elements (1344)

  // ---- load descriptor: strided src tile -> LDS ----
  u32x4 g0l;
  g0l[0] = 1u;                                            // count=1, valid
  g0l[1] = lds;                                           // lds_addr
  g0l[2] = (unsigned int)(srcAddr & 0xFFFFFFFFull);       // global_addr[31:0]
  g0l[3] = (unsigned int)((srcAddr >> 32) & 0x1FFFFFFull) // global_addr[56:32]
         | (2u << 30);                                    // type = 2 ("image")
  i32x8 g1l;
  g1l[0] = (int)(2u << 16);                               // data_size = 4B
  g1l[1] = (int)((SS & 0xFFFFu) << 16);                   // tensor_dim0[15:0]
  g1l[2] = (int)(((SS >> 16) & 0xFFFFu) | ((H & 0xFFFFu) << 16)); // td0 hi | td1 lo
  g1l[3] = (int)(((H >> 16) & 0xFFFFu) | (W << 16));      // td1 hi | tile_dim0
  g1l[4] = (int)H;                                        // tile_dim1 | tile_dim2=0
  g1l[5] = (int)SS;                                       // tensor_dim0_stride lo32
  g1l[6] = 0;                                             // stride0 hi16 | stride1 lo16
  g1l[7] = 0;                                             // stride1 hi32
  i32x4 gz4 = {0, 0, 0, 0};
  i32x8 gz8 = {0, 0, 0, 0, 0, 0, 0, 0};

  __builtin_amdgcn_tensor_load_to_lds(g0l, g1l, gz4, gz4, gz8, 0);
  __builtin_amdgcn_s_wait_tensorcnt(0);

  // ---- store descriptor: LDS -> contiguous dst tile ----
  u32x4 g0s;
  g0s[0] = 1u;
  g0s[1] = lds;
  g0s[2] = (unsigned int)(dstAddr & 0xFFFFFFFFull);
  g0s[3] = (unsigned int)((dstAddr >> 32) & 0x1FFFFFFull) | (2u << 30);
  i32x8 g1s;
  g1s[0] = (int)(2u << 16);
  g1s[1] = (int)((W & 0xFFFFu) << 16);                    // tensor_dim0 = 192
  g1s[2] = (int)(((W >> 16) & 0xFFFFu) | ((H & 0xFFFFu) << 16));
  g1s[3] = (int)(((H >> 16) & 0xFFFFu) | (W << 16));
  g1s[4] = (int)H;
  g1s[5] = (int)W;                                        // dst row stride = 192
  g1s[6] = 0;
  g1s[7] = 0;

  __builtin_amdgcn_tensor_store_from_lds(g0s, g1s, gz4, gz4, gz8, 0);
  __builtin_amdgcn_s_wait_tensorcnt(0);
  (void)buf;
#else
  // Fallback: direct float4 copy, 3072 float4 per patch, 96 per lane.
  const float4* src4 = (const float4*)(uintptr_t)srcAddr;  // row-strided
  float4* dst4 = (float4*)(uintptr_t)dstAddr;
  for (int i = threadIdx.x; i < PATCH * 48; i += 32) {
    int r = i / 48, q = i % 48;
    dst4[i] = *(const float4*)((const float*)src4 + (size_t)r * DIM * 3 + q * 4);
  }
  (void)buf;
#endif
}

// ---------------------------------------------------------------------------
extern "C" void kernel_launch(void* const* d_in, const int* in_sizes, int n_in,
                              void* d_out, int out_size, void* d_ws, size_t ws_size,
                              hipStream_t stream) {
  const float* heatmap = (const float*)d_in[0];  // [64,448,448,1] f32
  const float* image   = (const float*)d_in[1];  // [64,448,448,3] f32
  float* out = (float*)d_out;                    // [64*21,64,64,3] f32

  float* avg = (float*)d_ws;                                        // 64*49 f32
  int*   idx = (int*)((char*)d_ws + BATCH * POOLS * sizeof(float)); // 64*21 i32

  pool_kernel<<<BATCH * POOLS, 256, 0, stream>>>(heatmap, avg);
  topk_kernel<<<BATCH, 64, 0, stream>>>(avg, idx);
  gather_kernel<<<BATCH * NPTCH, 32, 0, stream>>>(image, idx, out);
}